// MultiPointPatchify_72241349919078
// MI455X (gfx1250) — compile-verified
//
#include <hip/hip_runtime.h>

#define FLT_MAX_C 3.402823466e+38f

typedef __attribute__((ext_vector_type(2))) float v2f;
typedef __attribute__((ext_vector_type(8))) float v8f;

__device__ __forceinline__ void argmax_combine(float& v, int& i, float ov, int oi) {
    if (ov > v || (ov == v && oi < i)) { v = ov; i = oi; }
}

// Issue async Global->LDS copies: each thread moves 16B per instruction,
// 128 threads * 16B = 2KB per round. Tracked on ASYNCcnt (6 ops/wave per 12KB slab).
__device__ __forceinline__ void async_slab_load(const float* gsrc, float* lds_dst,
                                                int tid, int bytes) {
    const uint64_t g = (uint64_t)(uintptr_t)gsrc;
    const unsigned l = (unsigned)(uintptr_t)lds_dst;   // generic LDS ptr low 32b == ds offset
#pragma unroll
    for (int off = 0; off < bytes; off += 128 * 16) {
        const unsigned la = l + off + tid * 16;
        const uint64_t ga = g + (uint64_t)(off + tid * 16);
        asm volatile("global_load_async_to_lds_b128 %0, %1, off"
                     :: "v"(la), "v"(ga)
                     : "memory");
    }
}

// ---------------------------------------------------------------------------
// Farthest point sampling: one block per batch. min_d lives in registers
// (PP floats per thread). Points re-read from L2 each of the NSEL iterations
// (384KB/batch vs 192MB L2) — the chain of dependent argmax reductions is the
// serial bottleneck of the whole op.
// ---------------------------------------------------------------------------
template<int NPTS, int NSEL, int NTHREADS>
__global__ __launch_bounds__(NTHREADS)
void fps_kernel(const float* __restrict__ pts, float* __restrict__ centers_out) {
    constexpr int PP     = NPTS / NTHREADS;
    constexpr int NWAVES = NTHREADS / 32;
    __shared__ int   s_sel;
    __shared__ float s_val[NWAVES];
    __shared__ int   s_idx[NWAVES];

    const int b    = blockIdx.x;
    const int tid  = threadIdx.x;
    const int lane = tid & 31;
    const int wid  = tid >> 5;
    const float* bp = pts + (size_t)b * NPTS * 3;

    float md[PP];
#pragma unroll
    for (int i = 0; i < PP; ++i) md[i] = FLT_MAX_C;

    int sel = 0;
    for (int k = 0; k < NSEL; ++k) {
        const float lx = bp[sel * 3 + 0];
        const float ly = bp[sel * 3 + 1];
        const float lz = bp[sel * 3 + 2];
        if (tid == 0) {
            float* c = centers_out + ((size_t)b * NSEL + k) * 3;
            c[0] = lx; c[1] = ly; c[2] = lz;
        }
        float best = -1.0f;
        int   bi   = 0;
#pragma unroll 4
        for (int i = 0; i < PP; ++i) {
            const int p = tid + i * NTHREADS;
            const float dx = bp[p * 3 + 0] - lx;
            const float dy = bp[p * 3 + 1] - ly;
            const float dz = bp[p * 3 + 2] - lz;
            const float d  = dx * dx + dy * dy + dz * dz;
            md[i] = fminf(md[i], d);
            if (md[i] > best) { best = md[i]; bi = p; }   // strict > keeps lowest index on ties
        }
        // wave32 argmax reduction
#pragma unroll
        for (int off = 16; off > 0; off >>= 1) {
            const float ov = __shfl_down(best, off, 32);
            const int   oi = __shfl_down(bi,   off, 32);
            argmax_combine(best, bi, ov, oi);
        }
        if (lane == 0) { s_val[wid] = best; s_idx[wid] = bi; }
        __syncthreads();
        if (wid == 0) {
            float v = (lane < NWAVES) ? s_val[lane] : -1.0f;
            int   i = (lane < NWAVES) ? s_idx[lane] : 0x7fffffff;
#pragma unroll
            for (int off = 16; off > 0; off >>= 1) {
                const float ov = __shfl_down(v, off, 32);
                const int   oi = __shfl_down(i, off, 32);
                argmax_combine(v, i, ov, oi);
            }
            if (lane == 0) s_sel = i;
        }
        __syncthreads();
        sel = s_sel;
    }
}

// ---------------------------------------------------------------------------
// KNN via V_WMMA_F32_16X16X4_F32 with async Global->LDS point staging.
//   score[m][n] = |p_n|^2 - 2 q_m . p_n   (ordering == true squared distance)
//   A (16x4):  row m = (qx, qy, qz, 1)        lanes 0-15 -> (K0,K1), 16-31 -> (K2,K3)
//   B (4x16):  col n = (-2px,-2py,-2pz,|p|^2) same half-lane split
// Points stream through a double-buffered 12KB LDS slab filled by
// global_load_async_to_lds_b128; the next slab's fill overlaps the
// latency-bound top-k insertion scan of the current one, and all 4 waves
// share one LDS copy instead of 4x re-reading L2.
// One wave owns 16 queries and a sorted top-KSEL insertion list in LDS.
// MODE 0: emit centered patches (float).  MODE 1: emit neighbor indices (int).
// ---------------------------------------------------------------------------
template<int KSEL, int MODE>
__global__ __launch_bounds__(128)
void knn_kernel(const float* __restrict__ query, const float* __restrict__ pts,
                int M, int N, int bpb,
                float* __restrict__ out_patch, int* __restrict__ out_idx) {
    constexpr int WPB  = 4;     // waves per block
    constexpr int SLAB = 1024;  // points per LDS slab (12KB)
    __shared__ float s_slab[2][SLAB * 3];
    __shared__ float s_d[WPB][16][KSEL];
    __shared__ int   s_i[WPB][16][KSEL];
    __shared__ float s_tile[WPB][16][17];   // +1 pad vs LDS banks

    const int b     = blockIdx.x / bpb;
    const int qb    = blockIdx.x % bpb;
    const int tid   = threadIdx.x;
    const int wave  = tid >> 5;
    const int lane  = tid & 31;
    const int qbase = qb * (WPB * 16) + wave * 16;

    const float* qbp = query + (size_t)b * M * 3;
    const float* pbp = pts   + (size_t)b * N * 3;

    // A fragment: built per-lane, no cross-lane traffic needed.
    const int   qm = qbase + (lane & 15);
    const float qx = qbp[qm * 3 + 0], qy = qbp[qm * 3 + 1], qz = qbp[qm * 3 + 2];
    v2f afrag;
    afrag[0] = (lane < 16) ? qx : qz;
    afrag[1] = (lane < 16) ? qy : 1.0f;

    if (lane < 16) {
        for (int j = 0; j < KSEL; ++j) { s_d[wave][lane][j] = FLT_MAX_C; s_i[wave][lane][j] = 0; }
    }

    const int nslabs = N / SLAB;
    async_slab_load(pbp, &s_slab[0][0], tid, SLAB * 12);          // prefetch slab 0

    for (int s = 0; s < nslabs; ++s) {
        const int cur = s & 1;
        if (s + 1 < nslabs) {
            async_slab_load(pbp + (size_t)(s + 1) * SLAB * 3, &s_slab[cur ^ 1][0],
                            tid, SLAB * 12);
            // 6 newest ops (next slab) may remain in flight; slab s is complete.
            asm volatile("s_wait_asynccnt 6" ::: "memory");
        } else {
            asm volatile("s_wait_asynccnt 0" ::: "memory");
        }
        __syncthreads();   // per-wave waits done -> slab visible to all waves

        const float* slab = &s_slab[cur][0];
        for (int c0 = 0; c0 < SLAB; c0 += 16) {
            const int   ln = c0 + (lane & 15);
            const float px = slab[ln * 3 + 0];
            const float py = slab[ln * 3 + 1];
            const float pz = slab[ln * 3 + 2];
            v2f bfrag;
            bfrag[0] = (lane < 16) ? (-2.0f * px) : (-2.0f * pz);
            bfrag[1] = (lane < 16) ? (-2.0f * py) : (px * px + py * py + pz * pz);

            v8f acc = {};
            acc = __builtin_amdgcn_wmma_f32_16x16x4_f32(
                    /*neg_a=*/false, afrag, /*neg_b=*/false, bfrag,
                    /*c_mod=*/(short)0, acc, /*reuse_a=*/false, /*reuse_b=*/false);

            __syncthreads();   // protect tile readers of previous chunk
            const int n  = lane & 15;
            const int mb = (lane >> 4) * 8;
#pragma unroll
            for (int r = 0; r < 8; ++r) s_tile[wave][mb + r][n] = acc[r];
            __syncthreads();

            if (lane < 16) {
                const int q = lane;
                const int gbase = s * SLAB + c0;
                float worst = s_d[wave][q][KSEL - 1];
#pragma unroll 1
                for (int j = 0; j < 16; ++j) {
                    const float d = s_tile[wave][q][j];
                    if (d < worst) {
                        int pos = KSEL - 1;
                        while (pos > 0 && s_d[wave][q][pos - 1] > d) {
                            s_d[wave][q][pos] = s_d[wave][q][pos - 1];
                            s_i[wave][q][pos] = s_i[wave][q][pos - 1];
                            --pos;
                        }
                        s_d[wave][q][pos] = d;
                        s_i[wave][q][pos] = gbase + j;
                        worst = s_d[wave][q][KSEL - 1];
                    }
                }
            }
        }
        __syncthreads();   // slab fully consumed before it is overwritten
    }
    __syncthreads();

    if (MODE == 0) {
        for (int q = 0; q < 16; ++q) {
            const int   gq = qbase + q;
            const float cx = qbp[gq * 3 + 0], cy = qbp[gq * 3 + 1], cz = qbp[gq * 3 + 2];
            for (int j = lane; j < KSEL; j += 32) {
                const int gi = s_i[wave][q][j];
                float* op = out_patch + (((size_t)b * M + gq) * KSEL + j) * 3;
                op[0] = pbp[gi * 3 + 0] - cx;
                op[1] = pbp[gi * 3 + 1] - cy;
                op[2] = pbp[gi * 3 + 2] - cz;
            }
        }
    } else {
        for (int q = 0; q < 16; ++q) {
            const int gq = qbase + q;
            for (int j = lane; j < KSEL; j += 32)
                out_idx[((size_t)b * M + gq) * KSEL + j] = s_i[wave][q][j];
        }
    }
}

// ---------------------------------------------------------------------------
// d_out layout (flat, reference return order):
//   patches  [4,2048,64,3] float   (1,572,864)
//   idx1     [4,64,32]     int32   (    8,192)
//   centers0 [4,2048,3]    float   (   24,576)
//   centers1 [4,64,3]      float   (      768)
// ---------------------------------------------------------------------------
extern "C" void kernel_launch(void* const* d_in, const int* in_sizes, int n_in,
                              void* d_out, int out_size, void* d_ws, size_t ws_size,
                              hipStream_t stream) {
    (void)in_sizes; (void)n_in; (void)out_size; (void)d_ws; (void)ws_size;

    constexpr int B = 4, N0 = 32768, M0 = 2048, K0 = 64, M1 = 64, K1 = 32;

    const float* points = (const float*)d_in[0];
    float* out      = (float*)d_out;
    float* patches  = out;                                           // B*M0*K0*3
    int*   idx1     = (int*)(out + (size_t)B * M0 * K0 * 3);         // B*M1*K1
    float* centers0 = out + (size_t)B * M0 * K0 * 3 + (size_t)B * M1 * K1;
    float* centers1 = centers0 + (size_t)B * M0 * 3;

    // Level 0: FPS 32768 -> 2048 centers
    fps_kernel<N0, M0, 1024><<<B, 1024, 0, stream>>>(points, centers0);

    // Level 0: KNN(k=64) + gather + center  -> patches
    knn_kernel<K0, 0><<<B * (M0 / 64), 128, 0, stream>>>(
        centers0, points, M0, N0, M0 / 64, patches, nullptr);

    // Level 1: FPS 2048 -> 64 centers
    fps_kernel<M0, M1, 256><<<B, 256, 0, stream>>>(centers0, centers1);

    // Level 1: KNN(k=32) indices only
    knn_kernel<K1, 1><<<B * (M1 / 64), 128, 0, stream>>>(
        centers1, centers0, M1, M0, M1 / 64, nullptr, idx1);
}